// GCN_50955492000289
// MI455X (gfx1250) — compile-verified
//
#include <hip/hip_runtime.h>
#include <math.h>

#define DF 64  // feature width of every GAT layer

typedef float v2f __attribute__((ext_vector_type(2)));
typedef float v8f __attribute__((ext_vector_type(8)));

// float atomic max via signed/unsigned integer ordering trick.
// m[] must be initialized to -inf (0xFF800000).
__device__ __forceinline__ float atomicMaxF(float* addr, float val) {
  if (val >= 0.f)
    return __int_as_float(atomicMax((int*)addr, __float_as_int(val)));
  return __uint_as_float(atomicMin((unsigned int*)addr, __float_as_uint(val)));
}

// ---------------------------------------------------------------------------
// GEMM: Hout[n,64] = H[n,K] @ Wm[K,64]  (+bias), optionally fused
// alpha_s[r] = Hout[r,:]·avec_s, alpha_d[r] = Hout[r,:]·avec_d.
// One 256-thread block = 8 waves; each wave owns a 16-row x 64-col tile and
// accumulates with V_WMMA_F32_16X16X4_F32 (full fp32 precision).
//
// Fragment layouts (CDNA5 ISA 7.12.2):
//   A 16x4 f32 : lane l<16  -> A[l][k0],A[l][k0+1] ; lane l>=16 -> A[l-16][k0+2..3]
//   B 4x16 f32 : lane l<16  -> B[k0][l],B[k0+1][l] ; lane l>=16 -> B[k0+2][l-16]...
//   C/D 16x16  : v8f, vgpr i = rows i (lanes 0-15) and i+8 (lanes 16-31)
//
// W is stored in LDS pre-swizzled into fragment order so a B fragment is one
// contiguous ds_load_b64 per lane (banks 0..63 covered exactly once):
//   frag f = k0 + t   (k0 multiple of 4, t = 0..3)
//   sW[f*64 + lane*2 + e] = Wm[(k0 + 2*(lane>>4) + e)*64 + t*16 + (lane&15)]
// ---------------------------------------------------------------------------
template<int K, bool ALPHA, bool BIAS>
__global__ __launch_bounds__(256)
void gat_gemm_kernel(const float* __restrict__ H, const float* __restrict__ Wm,
                     const float* __restrict__ bias,
                     const float* __restrict__ avec_s, const float* __restrict__ avec_d,
                     float* __restrict__ Hout,
                     float* __restrict__ alpha_s, float* __restrict__ alpha_d,
                     int n) {
  __shared__ __align__(16) float sW[K * DF];      // W in WMMA-fragment order
  __shared__ float sT[ALPHA ? 8 * 16 * DF : 8];   // per-wave 16x64 out staging

  const int tid  = threadIdx.x;
  const int wave = tid >> 5;
  const int lane = tid & 31;
  const int half = lane >> 4;
  const int l16  = lane & 15;

  // cooperative swizzled load of W into LDS (fragment order)
  for (int idx = tid; idx < K * DF; idx += 256) {
    const int f  = idx >> 6;       // fragment = k0 + t
    const int r  = idx & 63;
    const int ln = r >> 1;         // target lane
    const int e  = r & 1;          // element within lane's v2f
    const int k0 = f & ~3;
    const int t  = f & 3;
    const int k  = k0 + 2 * (ln >> 4) + e;
    const int c  = t * 16 + (ln & 15);
    sW[idx] = Wm[k * DF + c];
  }
  __syncthreads();

  const int rowbase = blockIdx.x * 128 + wave * 16;
  const int arow    = rowbase + l16;
  const int arowc   = arow < n ? arow : 0;            // clamp, keep EXEC full
  const float* Arow = H + (size_t)arowc * K;

  v8f acc[4] = {};

#pragma unroll
  for (int k0 = 0; k0 < K; k0 += 4) {
    const float2 av = *(const float2*)(Arow + k0 + 2 * half);
    v2f a; a.x = av.x; a.y = av.y;
#pragma unroll
    for (int t = 0; t < 4; ++t) {
      const v2f b = *(const v2f*)&sW[(k0 + t) * 64 + lane * 2];
      acc[t] = __builtin_amdgcn_wmma_f32_16x16x4_f32(
          false, a, false, b, (short)0, acc[t], false, false);
    }
  }

  // write out tile (+bias); stage into LDS for fused alpha dot products
#pragma unroll
  for (int t = 0; t < 4; ++t) {
#pragma unroll
    for (int i = 0; i < 8; ++i) {
      const int r   = i + 8 * half;
      const int row = rowbase + r;
      const int col = t * 16 + l16;
      float v = acc[t][i];
      if (BIAS) v += bias[col];
      if (ALPHA) sT[(wave * 16 + r) * DF + col] = v;
      if (row < n) Hout[(size_t)row * DF + col] = v;
    }
  }

  if (ALPHA) {
    __syncthreads();  // uniform (template const)
    // 32 lanes -> 16 rows x {alpha_s, alpha_d}; rotated column order avoids
    // LDS bank conflicts (bank = c, distinct across lanes).
    const float* av   = half ? avec_d : avec_s;
    const float* trow = sT + (wave * 16 + l16) * DF;
    float s = 0.f;
#pragma unroll
    for (int it = 0; it < DF; ++it) {
      const int c = (it + lane) & (DF - 1);
      s += trow[c] * av[c];
    }
    const int row = rowbase + l16;
    if (row < n) {
      if (half) alpha_d[row] = s; else alpha_s[row] = s;
    }
  }
}

// ---------------------------------------------------------------------------
// per-layer init: out accumulator = 0, m = -inf, denom = 0
// ---------------------------------------------------------------------------
__global__ void k_init(float* __restrict__ Hc, float* __restrict__ m,
                       float* __restrict__ denom, int n) {
  const int i = blockIdx.x * blockDim.x + threadIdx.x;
  if (i < n * DF) Hc[i] = 0.f;
  if (i < n) { m[i] = -INFINITY; denom[i] = 0.f; }
}

// edge pass 1: e = leaky_relu(a_s[src]+a_d[dst]); segment max over dst
__global__ void k_edge_e(const int* __restrict__ src, const int* __restrict__ dst,
                         const float* __restrict__ as, const float* __restrict__ ad,
                         float* __restrict__ ebuf, float* __restrict__ m,
                         int nE, int nN) {
  const int i = blockIdx.x * blockDim.x + threadIdx.x;
  if (i >= nE + nN) return;
  int s, d;
  if (i < nE) { s = src[i]; d = dst[i]; } else { s = i - nE; d = s; }
  float e = as[s] + ad[d];
  e = e > 0.f ? e : 0.2f * e;   // NEG_SLOPE = 0.2
  ebuf[i] = e;
  atomicMaxF(&m[d], e);
}

// edge pass 2: ex = exp(e - m[dst]); segment sum over dst
__global__ void k_edge_exp(const int* __restrict__ src, const int* __restrict__ dst,
                           float* __restrict__ ebuf, const float* __restrict__ m,
                           float* __restrict__ denom, int nE, int nN) {
  const int i = blockIdx.x * blockDim.x + threadIdx.x;
  if (i >= nE + nN) return;
  const int d = (i < nE) ? dst[i] : (i - nE);
  const float ex = __expf(ebuf[i] - m[d]);
  ebuf[i] = ex;
  atomicAdd(&denom[d], ex);
}

// edge pass 3: out[dst] += (ex/denom[dst]) * h[src]; 16 lanes per edge, float4
__global__ void k_edge_scatter(const int* __restrict__ src, const int* __restrict__ dst,
                               const float* __restrict__ ebuf, const float* __restrict__ denom,
                               const float* __restrict__ Hb, float* __restrict__ Hc,
                               int nE, int nN) {
  const long t = (long)blockIdx.x * blockDim.x + threadIdx.x;
  const int  i = (int)(t >> 4);
  if (i >= nE + nN) return;
  const int c = (int)(t & 15) << 2;
  int s, d;
  if (i < nE) { s = src[i]; d = dst[i]; } else { s = i - nE; d = s; }
  const float w = ebuf[i] / denom[d];
  const float4 hv = *(const float4*)(Hb + (size_t)s * DF + c);
  float* o = Hc + (size_t)d * DF + c;
  atomicAdd(o + 0, w * hv.x);
  atomicAdd(o + 1, w * hv.y);
  atomicAdd(o + 2, w * hv.z);
  atomicAdd(o + 3, w * hv.w);
}

// node finalize: h = (relu?) (out + b)
__global__ void k_finalize(const float* __restrict__ Hc, const float* __restrict__ b,
                           int relu, float* __restrict__ Hout, int n) {
  const int i = blockIdx.x * blockDim.x + threadIdx.x;
  if (i >= n * DF) return;
  float v = Hc[i] + b[i & (DF - 1)];
  Hout[i] = relu ? fmaxf(v, 0.f) : v;
}

// L2 normalize each row; 16 lanes per node, float4 + shuffle reduction
__global__ void k_normalize(const float* __restrict__ Hin, float* __restrict__ out, int n) {
  const long t    = (long)blockIdx.x * blockDim.x + threadIdx.x;
  const int  node = (int)(t >> 4);
  if (node >= n) return;
  const int c = (int)(t & 15) << 2;
  const float4 v = *(const float4*)(Hin + (size_t)node * DF + c);
  float ss = v.x * v.x + v.y * v.y + v.z * v.z + v.w * v.w;
  ss += __shfl_xor(ss, 1, 16);
  ss += __shfl_xor(ss, 2, 16);
  ss += __shfl_xor(ss, 4, 16);
  ss += __shfl_xor(ss, 8, 16);
  const float sc = 1.f / fmaxf(sqrtf(ss), 1e-12f);
  float4 o; o.x = v.x * sc; o.y = v.y * sc; o.z = v.z * sc; o.w = v.w * sc;
  *(float4*)(out + (size_t)node * DF + c) = o;
}

// ---------------------------------------------------------------------------
extern "C" void kernel_launch(void* const* d_in, const int* in_sizes, int n_in,
                              void* d_out, int out_size, void* d_ws, size_t ws_size,
                              hipStream_t stream) {
  const float* x    = (const float*)d_in[0];
  const int*   esrc = (const int*)d_in[1];
  const int*   edst = (const int*)d_in[2];
  const float* Wpre = (const float*)d_in[3];
  const float* bpre = (const float*)d_in[4];
  const float* W1   = (const float*)d_in[5];
  const float* as1  = (const float*)d_in[6];
  const float* ad1  = (const float*)d_in[7];
  const float* b1   = (const float*)d_in[8];
  const float* Wh   = (const float*)d_in[9];
  const float* ash  = (const float*)d_in[10];
  const float* adh  = (const float*)d_in[11];
  const float* bh   = (const float*)d_in[12];
  const float* Wo   = (const float*)d_in[13];
  const float* aso  = (const float*)d_in[14];
  const float* ado  = (const float*)d_in[15];
  const float* bo   = (const float*)d_in[16];

  const int N  = in_sizes[0] / 128;
  const int E  = in_sizes[1];
  const int ET = E + N;  // edges incl. self loops

  float* ws = (float*)d_ws;
  float* Ha = ws;      ws += (size_t)N * DF;  // current node features
  float* Hb = ws;      ws += (size_t)N * DF;  // h = Ha @ W
  float* Hc = ws;      ws += (size_t)N * DF;  // scatter accumulator
  float* asb = ws;     ws += N;               // alpha_s per node
  float* adb = ws;     ws += N;               // alpha_d per node
  float* mb  = ws;     ws += N;               // segment max
  float* db  = ws;     ws += N;               // segment sum (denom)
  float* eb  = ws;     ws += ET;              // per-edge e / ex

  const dim3 blk(256);
  const dim3 g_gemm((N + 127) / 128);
  const dim3 g_nv((N * DF + 255) / 256);
  const dim3 g_e((ET + 255) / 256);
  const dim3 g_sc((unsigned)(((long)ET * 16 + 255) / 256));
  const dim3 g_nm((unsigned)(((long)N * 16 + 255) / 256));

  // linear_pre: Ha = x @ Wpre + bpre   (K=128, bias, no alpha)
  gat_gemm_kernel<128, false, true><<<g_gemm, blk, 0, stream>>>(
      x, Wpre, bpre, nullptr, nullptr, Ha, nullptr, nullptr, N);

  auto gat_layer = [&](const float* W, const float* as, const float* ad,
                       const float* b, int relu, float* hout) {
    gat_gemm_kernel<64, true, false><<<g_gemm, blk, 0, stream>>>(
        Ha, W, nullptr, as, ad, Hb, asb, adb, N);
    k_init<<<g_nv, blk, 0, stream>>>(Hc, mb, db, N);
    k_edge_e<<<g_e, blk, 0, stream>>>(esrc, edst, asb, adb, eb, mb, E, N);
    k_edge_exp<<<g_e, blk, 0, stream>>>(esrc, edst, eb, mb, db, E, N);
    k_edge_scatter<<<g_sc, blk, 0, stream>>>(esrc, edst, eb, db, Hb, Hc, E, N);
    k_finalize<<<g_nv, blk, 0, stream>>>(Hc, b, relu, hout, N);
  };

  gat_layer(W1, as1, ad1, b1, 1, Ha);
  for (int i = 0; i < 6; ++i)
    gat_layer(Wh + (size_t)i * DF * DF, ash + i * DF, adh + i * DF, bh + i * DF, 1, Ha);
  gat_layer(Wo, aso, ado, bo, 0, Hb);   // final layer: no relu, result in Hb

  k_normalize<<<g_nm, blk, 0, stream>>>(Hb, (float*)d_out, N);
}